// Seq2Seq_87600152969727
// MI455X (gfx1250) — compile-verified
//
#include <hip/hip_runtime.h>
#include <stdint.h>

typedef __attribute__((ext_vector_type(2))) float v2f;
typedef __attribute__((ext_vector_type(8))) float v8f;

#define VOCAB 50257
#define EMB   32
#define HID   32
#define S_SRC 64
#define S_TGT 64
#define BATCH 32

__device__ __forceinline__ float sigm_(float x) { return 1.0f / (1.0f + __expf(-x)); }

// Load a 16x4 (A) or 4x16-transposed (B) f32 WMMA fragment for one k-step.
// Source is row-major [rows, 32]; per 16x16x4 f32 WMMA the per-lane mapping is:
//   lanes 0-15  : row = base + lane,      K = 4*kk + {0,1}
//   lanes 16-31 : row = base + (lane-16), K = 4*kk + {2,3}
// i.e. one aligned float2 load per lane.
__device__ __forceinline__ v2f ldfrag(const float* p, int row, int kk, int csel) {
    return *(const v2f*)(p + (size_t)row * 32 + kk * 4 + csel);
}

// One LSTM cell for the whole 256-thread block (8 waves).
// xbuf/hbuf/cbuf: [32][32] row-major (LDS). g: [32][128] scratch (LDS).
// Gates = bias + x@Wih^T + h@Whh^T via v_wmma_f32_16x16x4_f32.
// Wave w owns gate columns [16w, 16w+16), both batch halves (2 accumulators).
__device__ void lstm_cell(const float* __restrict__ xbuf, float* hbuf, float* cbuf,
                          const float* __restrict__ Wih, const float* __restrict__ Whh,
                          const float* __restrict__ bih, const float* __restrict__ bhh,
                          float* g) {
    const int tid = threadIdx.x;
    const int wave = tid >> 5, lane = tid & 31;
    const int lanecol = lane & 15;
    const int csel = (lane & 16) ? 2 : 0;
    const int n = wave * 16 + lanecol;

    const float bias = bih[n] + bhh[n];
    v8f acc0, acc1;
#pragma unroll
    for (int r = 0; r < 8; ++r) { acc0[r] = bias; acc1[r] = bias; }

#pragma unroll
    for (int kk = 0; kk < 8; ++kk) {             // K = EMB/HID = 32
        v2f bf = ldfrag(Wih, n, kk, csel);
        v2f a0 = ldfrag(xbuf, lanecol, kk, csel);
        v2f a1 = ldfrag(xbuf, 16 + lanecol, kk, csel);
        acc0 = __builtin_amdgcn_wmma_f32_16x16x4_f32(false, a0, false, bf, (short)0, acc0, false, false);
        acc1 = __builtin_amdgcn_wmma_f32_16x16x4_f32(false, a1, false, bf, (short)0, acc1, false, false);
    }
#pragma unroll
    for (int kk = 0; kk < 8; ++kk) {             // + h @ Whh^T
        v2f bf = ldfrag(Whh, n, kk, csel);
        v2f a0 = ldfrag(hbuf, lanecol, kk, csel);
        v2f a1 = ldfrag(hbuf, 16 + lanecol, kk, csel);
        acc0 = __builtin_amdgcn_wmma_f32_16x16x4_f32(false, a0, false, bf, (short)0, acc0, false, false);
        acc1 = __builtin_amdgcn_wmma_f32_16x16x4_f32(false, a1, false, bf, (short)0, acc1, false, false);
    }

    // C layout: VGPR r, lanes 0-15 -> M=r, lanes 16-31 -> M=8+r
    const int rbase = (lane & 16) ? 8 : 0;
#pragma unroll
    for (int r = 0; r < 8; ++r) {
        g[(rbase + r) * 128 + n]      = acc0[r];
        g[(16 + rbase + r) * 128 + n] = acc1[r];
    }
    __syncthreads();

    // Pointwise gate math: i,f,g,o at columns j, 32+j, 64+j, 96+j
#pragma unroll
    for (int e = 0; e < 4; ++e) {
        int idx = tid + 256 * e;
        int bb = idx >> 5, j = idx & 31;
        float gi = g[bb * 128 + j];
        float gf = g[bb * 128 + 32 + j];
        float gg = g[bb * 128 + 64 + j];
        float go = g[bb * 128 + 96 + j];
        float c2 = sigm_(gf) * cbuf[idx] + sigm_(gi) * tanhf(gg);
        cbuf[idx] = c2;
        hbuf[idx] = sigm_(go) * tanhf(c2);
    }
    __syncthreads();
}

// ----- encoder: both layers fused per timestep, single block -----
__global__ void __launch_bounds__(256)
encoder_kernel(const int* __restrict__ x_src, const float* __restrict__ emb,
               const float* W0i, const float* W0h, const float* b0i, const float* b0h,
               const float* W1i, const float* W1h, const float* b1i, const float* b1h,
               float* __restrict__ state) {
    __shared__ __align__(16) float xbuf[1024], h0[1024], c0b[1024], h1[1024], c1b[1024];
    __shared__ __align__(16) float g[32 * 128];
    __shared__ int toks[32];
    const int tid = threadIdx.x;
    for (int i = tid; i < 1024; i += 256) { h0[i] = 0.f; c0b[i] = 0.f; h1[i] = 0.f; c1b[i] = 0.f; }
    __syncthreads();
    for (int t = 0; t < S_SRC; ++t) {
        if (tid < 32) toks[tid] = x_src[t * BATCH + tid];
        __syncthreads();
        for (int i = tid; i < 1024; i += 256) {
            int bb = i >> 5, j = i & 31;
            xbuf[i] = emb[(size_t)toks[bb] * EMB + j];
        }
        __syncthreads();
        lstm_cell(xbuf, h0, c0b, W0i, W0h, b0i, b0h, g);
        lstm_cell(h0,   h1, c1b, W1i, W1h, b1i, b1h, g);
    }
    for (int i = tid; i < 1024; i += 256) {
        state[i] = h0[i]; state[1024 + i] = c0b[i];
        state[2048 + i] = h1[i]; state[3072 + i] = c1b[i];
    }
}

// ----- argmax key packing: monotone f32 encoding | ~col (first-max tiebreak) -----
__device__ __forceinline__ unsigned long long pack_key(float v, unsigned int n) {
    unsigned int b = __float_as_uint(v);
    unsigned int e = (b & 0x80000000u) ? ~b : (b | 0x80000000u);
    return ((unsigned long long)e << 32) | (unsigned long long)(~n);
}

__global__ void init_kernel(unsigned long long* packed, int n) {
    int i = blockIdx.x * blockDim.x + threadIdx.x;
    if (i < n) packed[i] = 0ULL;
}

// ----- decoder cell step: token decode -> embed -> 2 LSTM cells -----
__global__ void __launch_bounds__(256)
dec_cell_kernel(int step, const int* __restrict__ x_tgt,
                const unsigned long long* __restrict__ packed,
                const float* __restrict__ emb,
                const float* W0i, const float* W0h, const float* b0i, const float* b0h,
                const float* W1i, const float* W1h, const float* b1i, const float* b1h,
                float* __restrict__ state) {
    __shared__ __align__(16) float xbuf[1024], h0[1024], c0b[1024], h1[1024], c1b[1024];
    __shared__ __align__(16) float g[32 * 128];
    __shared__ int toks[32];
    const int tid = threadIdx.x;
    for (int i = tid; i < 1024; i += 256) {
        h0[i]  = state[i];        c0b[i] = state[1024 + i];
        h1[i]  = state[2048 + i]; c1b[i] = state[3072 + i];
    }
    if (tid < 32) {
        int tok;
        if (step == 0) tok = x_tgt[tid];                       // x_tgt[0][b]
        else {
            unsigned long long key = packed[(size_t)(step - 1) * 32 + tid];
            tok = (int)(~(unsigned int)(key & 0xFFFFFFFFu));
        }
        toks[tid] = tok;
    }
    __syncthreads();
    for (int i = tid; i < 1024; i += 256) {
        int bb = i >> 5, j = i & 31;
        xbuf[i] = emb[(size_t)toks[bb] * EMB + j];
    }
    __syncthreads();
    lstm_cell(xbuf, h0, c0b, W0i, W0h, b0i, b0h, g);
    lstm_cell(h0,   h1, c1b, W1i, W1h, b1i, b1h, g);
    for (int i = tid; i < 1024; i += 256) {
        state[i] = h0[i]; state[1024 + i] = c0b[i];
        state[2048 + i] = h1[i]; state[3072 + i] = c1b[i];
    }
}

// ----- fc + fused argmax: logits = h1 @ fcW^T + fcb, wave = one 16-col tile -----
// 405 MB of logits are write-once / never-read: store them non-temporal (TH=NT)
// so the streaming writes don't evict fc_W (6.4 MB, re-read 63x) from L2.
__global__ void __launch_bounds__(256)
fc_kernel(int step, const float* __restrict__ state,
          const float* __restrict__ fcW, const float* __restrict__ fcb,
          float* __restrict__ out, unsigned long long* __restrict__ packed) {
    const float* h1 = state + 2048;
    __shared__ unsigned long long lmax[32];
    const int tid = threadIdx.x;
    if (tid < 32) lmax[tid] = 0ULL;
    __syncthreads();

    const int wave = tid >> 5, lane = tid & 31;
    const int lanecol = lane & 15;
    const int csel = (lane & 16) ? 2 : 0;
    const int n0 = (blockIdx.x * 8 + wave) * 16;

    if (n0 < VOCAB) {
        const int nn = n0 + lanecol;
        const int nc = nn < VOCAB ? nn : VOCAB - 1;
        const bool valid = nn < VOCAB;

        const float bias = fcb[nc];
        v8f acc0, acc1;
#pragma unroll
        for (int r = 0; r < 8; ++r) { acc0[r] = bias; acc1[r] = bias; }

#pragma unroll
        for (int kk = 0; kk < 8; ++kk) {         // K = HID = 32
            v2f bf = ldfrag(fcW, nc, kk, csel);  // B tile: fcW rows = vocab cols
            v2f a0 = ldfrag(h1, lanecol, kk, csel);
            v2f a1 = ldfrag(h1, 16 + lanecol, kk, csel);
            acc0 = __builtin_amdgcn_wmma_f32_16x16x4_f32(false, a0, false, bf, (short)0, acc0, false, false);
            acc1 = __builtin_amdgcn_wmma_f32_16x16x4_f32(false, a1, false, bf, (short)0, acc1, false, false);
        }

        float* orow = out + (size_t)step * BATCH * (size_t)VOCAB;
        const int rbase = (lane & 16) ? 8 : 0;
#pragma unroll
        for (int r = 0; r < 8; ++r) {
            const int row0 = rbase + r, row1 = rbase + r + 16;
            const float v0 = acc0[r], v1 = acc1[r];
            if (valid) {
                __builtin_nontemporal_store(v0, &orow[(size_t)row0 * VOCAB + nn]);
                __builtin_nontemporal_store(v1, &orow[(size_t)row1 * VOCAB + nn]);
            }
            unsigned long long k0 = pack_key(valid ? v0 : -INFINITY, (unsigned)nn);
            unsigned long long k1 = pack_key(valid ? v1 : -INFINITY, (unsigned)nn);
#pragma unroll
            for (int off = 1; off < 16; off <<= 1) {   // reduce within 16-lane groups
                unsigned long long o0 = __shfl_xor(k0, off, 16);
                unsigned long long o1 = __shfl_xor(k1, off, 16);
                k0 = k0 > o0 ? k0 : o0;
                k1 = k1 > o1 ? k1 : o1;
            }
            if (lanecol == 0) {
                atomicMax(&lmax[row0], k0);
                atomicMax(&lmax[row1], k1);
            }
        }
    }
    __syncthreads();
    if (tid < 32) {
        unsigned long long v = lmax[tid];
        if (v) atomicMax(&packed[(size_t)step * 32 + tid], v);
    }
}

extern "C" void kernel_launch(void* const* d_in, const int* in_sizes, int n_in,
                              void* d_out, int out_size, void* d_ws, size_t ws_size,
                              hipStream_t stream) {
    const int*   x_src   = (const int*)d_in[0];
    const int*   x_tgt   = (const int*)d_in[1];
    const float* enc_emb = (const float*)d_in[2];
    const float *eW0i = (const float*)d_in[3],  *eW0h = (const float*)d_in[4];
    const float *eb0i = (const float*)d_in[5],  *eb0h = (const float*)d_in[6];
    const float *eW1i = (const float*)d_in[7],  *eW1h = (const float*)d_in[8];
    const float *eb1i = (const float*)d_in[9],  *eb1h = (const float*)d_in[10];
    const float* dec_emb = (const float*)d_in[11];
    const float *dW0i = (const float*)d_in[12], *dW0h = (const float*)d_in[13];
    const float *db0i = (const float*)d_in[14], *db0h = (const float*)d_in[15];
    const float *dW1i = (const float*)d_in[16], *dW1h = (const float*)d_in[17];
    const float *db1i = (const float*)d_in[18], *db1h = (const float*)d_in[19];
    const float* fcW = (const float*)d_in[20];
    const float* fcb = (const float*)d_in[21];

    float* out   = (float*)d_out;
    float* state = (float*)d_ws;                                   // 4 x 1024 f32
    unsigned long long* packed = (unsigned long long*)((char*)d_ws + 16384);

    const int NSTEP = S_TGT - 1;                                   // 63
    init_kernel<<<(NSTEP * 32 + 255) / 256, 256, 0, stream>>>(packed, NSTEP * 32);
    encoder_kernel<<<1, 256, 0, stream>>>(x_src, enc_emb,
                                          eW0i, eW0h, eb0i, eb0h,
                                          eW1i, eW1h, eb1i, eb1h, state);
    const int tiles = (VOCAB + 15) / 16;        // 3142
    const int fcblocks = (tiles + 7) / 8;       // 393 blocks x 8 waves
    for (int s = 0; s < NSTEP; ++s) {
        dec_cell_kernel<<<1, 256, 0, stream>>>(s, x_tgt, packed, dec_emb,
                                               dW0i, dW0h, db0i, db0h,
                                               dW1i, dW1h, db1i, db1h, state);
        fc_kernel<<<fcblocks, 256, 0, stream>>>(s, state, fcW, fcb, out, packed);
    }
}